// Detection_51024211476842
// MI455X (gfx1250) — compile-verified
//
#include <hip/hip_runtime.h>
#include <math.h>

// ---------------------------------------------------------------------------
// MI455X (gfx1250) detection pipeline.
//   - pointwise (1x1) convs + head as fp32 WMMA GEMMs (v_wmma_f32_16x16x4_f32),
//     K fully unrolled at compile time so loads pipeline ahead of the WMMA chain
//   - depthwise 3x3 stride-2 convs as VALU (K=1 per channel, no matrix shape)
//   - decode + greedy NMS fused into a single workgroup (serial dependency)
// All intermediates live in d_ws (NHWC), trivially resident in 192MB L2.
// ---------------------------------------------------------------------------

typedef float v2f __attribute__((ext_vector_type(2)));
typedef float v8f __attribute__((ext_vector_type(8)));

#define BN_EPS 1e-5f

__device__ __forceinline__ float relu6f(float x) {
    return fminf(fmaxf(x, 0.0f), 6.0f);
}

// ---------------------------------------------------------------------------
// Block 0 fused: DW 3x3 s2 on the single input channel -> BN -> ReLU6 ->
// PW 1x1 (1->16) -> BN -> ReLU6.  Output NHWC [128,128,16].
// ---------------------------------------------------------------------------
__global__ void block0_kernel(const float* __restrict__ x,
                              const float* __restrict__ dw_w,
                              const float* __restrict__ g1, const float* __restrict__ b1,
                              const float* __restrict__ m1, const float* __restrict__ v1,
                              const float* __restrict__ pw_w,
                              const float* __restrict__ g2, const float* __restrict__ b2,
                              const float* __restrict__ m2, const float* __restrict__ v2,
                              float* __restrict__ out) {
    int idx = blockIdx.x * blockDim.x + threadIdx.x;
    if (idx >= 128 * 128) return;
    int ox = idx & 127, oy = idx >> 7;

    float s = 0.0f;
#pragma unroll
    for (int ky = 0; ky < 3; ++ky) {
#pragma unroll
        for (int kx = 0; kx < 3; ++kx) {
            int iy = 2 * oy - 1 + ky;
            int ix = 2 * ox - 1 + kx;
            if (iy >= 0 && iy < 256 && ix >= 0 && ix < 256)
                s += x[iy * 256 + ix] * dw_w[ky * 3 + kx];
        }
    }
    float sc1 = g1[0] * rsqrtf(v1[0] + BN_EPS);
    float sh1 = b1[0] - m1[0] * sc1;
    float h = relu6f(s * sc1 + sh1);

    float* o = out + (size_t)idx * 16;
#pragma unroll
    for (int co = 0; co < 16; ++co) {
        float sc = g2[co] * rsqrtf(v2[co] + BN_EPS);
        float sh = b2[co] - m2[co] * sc;
        o[co] = relu6f(h * pw_w[co] * sc + sh);
    }
}

// ---------------------------------------------------------------------------
// Generic depthwise 3x3 stride-2 + BN + ReLU6, NHWC -> NHWC (H/2, W/2).
// One thread per output element; adjacent lanes cover adjacent channels.
// ---------------------------------------------------------------------------
__global__ void dw_kernel(const float* __restrict__ in,
                          const float* __restrict__ w,
                          const float* __restrict__ g, const float* __restrict__ b,
                          const float* __restrict__ m, const float* __restrict__ v,
                          float* __restrict__ out, int C, int Hin, int Win) {
    int Ho = Hin >> 1, Wo = Win >> 1;
    int total = C * Ho * Wo;
    int idx = blockIdx.x * blockDim.x + threadIdx.x;
    if (idx >= total) return;
    int c = idx % C;
    int p = idx / C;
    int ox = p % Wo, oy = p / Wo;

    const float* wc = w + c * 9;
    float s = 0.0f;
#pragma unroll
    for (int ky = 0; ky < 3; ++ky) {
#pragma unroll
        for (int kx = 0; kx < 3; ++kx) {
            int iy = 2 * oy - 1 + ky;
            int ix = 2 * ox - 1 + kx;
            if (iy >= 0 && iy < Hin && ix >= 0 && ix < Win)
                s += in[((size_t)iy * Win + ix) * C + c] * wc[ky * 3 + kx];
        }
    }
    float sc = g[c] * rsqrtf(v[c] + BN_EPS);
    float sh = b[c] - m[c] * sc;
    out[(size_t)p * C + c] = relu6f(s * sc + sh);
}

// ---------------------------------------------------------------------------
// Pointwise 1x1 conv as GEMM:  out[M,N] = A[M,K] * W^T  (W is [N=co, K=ci]),
// fused BN + ReLU6.  One wave computes one 16x16 output tile with
// V_WMMA_F32_16X16X4_F32 (fp32 A/B, fp32 accumulate).
// A-frag (16x4 f32): lane<16 -> M=lane, VGPR j = K(khalf+j); lanes 16-31 hold K+2.
// B-frag mirrored (lane&15 = N).  D: VGPR v -> M = v + 8*(lane>=16), N = lane&15.
// M/K/N compile-time: K-loop fully unrolls so the b64 fragment loads issue
// ahead of the WMMA chain; tile div/mod are shifts.
// Tile counts are exact multiples of waves -> EXEC all-ones for WMMA.
// ---------------------------------------------------------------------------
template <int M, int K, int N>
__global__ void pw_gemm_wmma(const float* __restrict__ A,
                             const float* __restrict__ W,
                             const float* __restrict__ bng, const float* __restrict__ bnb,
                             const float* __restrict__ bnm, const float* __restrict__ bnv,
                             float* __restrict__ out) {
    constexpr int tilesN = N / 16;
    constexpr int totalTiles = (M / 16) * tilesN;

    int wave = (blockIdx.x * blockDim.x + threadIdx.x) >> 5;
    if (wave >= totalTiles) return;  // wave-uniform; never splits a wave here
    int lane = threadIdx.x & 31;
    int tm = wave / tilesN;
    int tn = wave % tilesN;
    int row = tm * 16 + (lane & 15);
    int col = tn * 16 + (lane & 15);
    int khalf = (lane >> 4) << 1;  // 0 or 2

    const float* arow = A + (size_t)row * K + khalf;
    const float* wrow = W + (size_t)col * K + khalf;

    // gfx1250 prefetch path: data is L2-resident (produced by previous kernel),
    // pull toward near cache (locality 3 -> WGP scope).
    __builtin_prefetch(arow, 0, 3);
    __builtin_prefetch(wrow, 0, 3);

    v8f acc = {};
#pragma unroll
    for (int k0 = 0; k0 < K; k0 += 4) {
        v2f a, b;
        a.x = arow[k0];
        a.y = arow[k0 + 1];
        b.x = wrow[k0];
        b.y = wrow[k0 + 1];
        acc = __builtin_amdgcn_wmma_f32_16x16x4_f32(false, a, false, b,
                                                    (short)0, acc, false, false);
    }

    float sc = bng[col] * rsqrtf(bnv[col] + BN_EPS);
    float sh = bnb[col] - bnm[col] * sc;
    int mb = tm * 16 + ((lane >> 4) << 3);
#pragma unroll
    for (int vv = 0; vv < 8; ++vv) {
        out[(size_t)(mb + vv) * N + col] = relu6f(acc[vv] * sc + sh);
    }
}

// ---------------------------------------------------------------------------
// Detection heads as one WMMA GEMM: M=256 pixels, K=128, N=16 (cols 0-3 = cls,
// 4-11 = reg, 12-15 pad).  Pad columns point at a VALID weight row (cls_w row 0)
// so the B-fragment load is uniform/coalesced (no divergent EXEC-masked loads);
// their outputs are junk but are never read downstream (decode uses chans 0-11).
// Bias added, no activation.  out: [256,16].
// ---------------------------------------------------------------------------
__global__ void head_gemm_wmma(const float* __restrict__ A,
                               const float* __restrict__ cls_w, const float* __restrict__ cls_b,
                               const float* __restrict__ reg_w, const float* __restrict__ reg_b,
                               float* __restrict__ out) {
    constexpr int K = 128;
    int wave = (blockIdx.x * blockDim.x + threadIdx.x) >> 5;
    if (wave >= 16) return;
    int lane = threadIdx.x & 31;
    int row = wave * 16 + (lane & 15);
    int col = lane & 15;
    int khalf = (lane >> 4) << 1;

    const float* arow = A + (size_t)row * K + khalf;
    const float* wrow = (col < 4)    ? cls_w + (size_t)col * K + khalf
                        : (col < 12) ? reg_w + (size_t)(col - 4) * K + khalf
                                     : cls_w + khalf;  // valid addr; result dead
    float bias = (col < 4) ? cls_b[col] : (col < 12) ? reg_b[col - 4] : 0.0f;

    __builtin_prefetch(arow, 0, 3);

    v8f acc = {};
#pragma unroll
    for (int k0 = 0; k0 < K; k0 += 4) {
        v2f a, b;
        a.x = arow[k0];
        a.y = arow[k0 + 1];
        b.x = wrow[k0];
        b.y = wrow[k0 + 1];
        acc = __builtin_amdgcn_wmma_f32_16x16x4_f32(false, a, false, b,
                                                    (short)0, acc, false, false);
    }

    int mb = wave * 16 + ((lane >> 4) << 3);
#pragma unroll
    for (int vv = 0; vv < 8; ++vv) {
        out[(size_t)(mb + vv) * 16 + col] = acc[vv] + bias;
    }
}

// ---------------------------------------------------------------------------
// Softmax + box decode + greedy NMS (100 iterations), one workgroup of 512
// threads (1 thread per anchor).  LDS tree argmax; ties -> lowest index
// (matches jnp.argmax).  Writes d_out[100,5].
// ---------------------------------------------------------------------------
#define NANCH 512

__global__ void decode_nms_kernel(const float* __restrict__ head,
                                  const float* __restrict__ anchors,
                                  float* __restrict__ out) {
    __shared__ float sx1[NANCH], sy1[NANCH], sx2[NANCH], sy2[NANCH];
    __shared__ float ss[NANCH];
    __shared__ float rv[NANCH];
    __shared__ int   ri[NANCH];

    int t = threadIdx.x;
    int pix = t >> 1, al = t & 1;
    const float* hr = head + (size_t)pix * 16;

    // class prob (softmax over 2 classes == sigmoid of logit diff)
    float c0 = hr[al * 2 + 0];
    float c1 = hr[al * 2 + 1];
    float p1 = 1.0f / (1.0f + expf(c0 - c1));

    // box decode
    float t0 = hr[4 + al * 4 + 0];
    float t1 = hr[4 + al * 4 + 1];
    float t2 = hr[4 + al * 4 + 2];
    float t3 = hr[4 + al * 4 + 3];
    float acx = anchors[t * 4 + 0];
    float acy = anchors[t * 4 + 1];
    float aw  = anchors[t * 4 + 2];
    float ah  = anchors[t * 4 + 3];
    float cx = t0 * aw + acx;
    float cy = t1 * ah + acy;
    float bw = expf(t2) * aw;
    float bh = expf(t3) * ah;
    sx1[t] = cx - bw * 0.5f;
    sy1[t] = cy - bh * 0.5f;
    sx2[t] = cx + bw * 0.5f;
    sy2[t] = cy + bh * 0.5f;
    ss[t] = (p1 > 0.5f) ? p1 : -1.0f;
    __syncthreads();

    float myArea = (sx2[t] - sx1[t]) * (sy2[t] - sy1[t]);

    for (int it = 0; it < 100; ++it) {
        rv[t] = ss[t];
        ri[t] = t;
        __syncthreads();
#pragma unroll
        for (int off = NANCH / 2; off > 0; off >>= 1) {
            if (t < off) {
                float ov = rv[t + off];
                int oi = ri[t + off];
                if (ov > rv[t] || (ov == rv[t] && oi < ri[t])) {
                    rv[t] = ov;
                    ri[t] = oi;
                }
            }
            __syncthreads();
        }
        int best = ri[0];
        float top = rv[0];
        float bx1 = sx1[best], by1 = sy1[best], bx2 = sx2[best], by2 = sy2[best];
        int valid = top > 0.0f;
        if (t == 0) {
            out[it * 5 + 0] = valid ? bx1 : 0.0f;
            out[it * 5 + 1] = valid ? by1 : 0.0f;
            out[it * 5 + 2] = valid ? bx2 : 0.0f;
            out[it * 5 + 3] = valid ? by2 : 0.0f;
            out[it * 5 + 4] = valid ? top : 0.0f;
        }
        // IoU of best vs my box; suppress (scores persist across iterations)
        float lx = fmaxf(bx1, sx1[t]);
        float ly = fmaxf(by1, sy1[t]);
        float rx = fminf(bx2, sx2[t]);
        float ry = fminf(by2, sy2[t]);
        float iw = fmaxf(rx - lx, 0.0f);
        float ih = fmaxf(ry - ly, 0.0f);
        float inter = iw * ih;
        float aB = (bx2 - bx1) * (by2 - by1);
        float iou = inter / (aB + myArea - inter + 1e-9f);
        if (iou > 0.5f || t == best) ss[t] = -INFINITY;
        __syncthreads();
    }
}

// ---------------------------------------------------------------------------
// Launch: 9 kernels on `stream`; intermediates in d_ws (NHWC, ~2.4 MB).
// Input order (setup_inputs insertion order): x, then 4 blocks of
// [dw_w, bn1.g,b,m,v, pw_w, bn2.g,b,m,v], then cls_w, cls_b, reg_w, reg_b,
// then anchors.
// ---------------------------------------------------------------------------
extern "C" void kernel_launch(void* const* d_in, const int* in_sizes, int n_in,
                              void* d_out, int out_size, void* d_ws, size_t ws_size,
                              hipStream_t stream) {
    (void)in_sizes; (void)n_in; (void)out_size; (void)ws_size;

    const float* x = (const float*)d_in[0];
#define BP(b, j) ((const float*)d_in[1 + (b) * 10 + (j)])
    const float* cls_w   = (const float*)d_in[41];
    const float* cls_b   = (const float*)d_in[42];
    const float* reg_w   = (const float*)d_in[43];
    const float* reg_b   = (const float*)d_in[44];
    const float* anchors = (const float*)d_in[45];

    float* ws    = (float*)d_ws;
    float* act0  = ws;               // 128*128*16 = 262144
    float* act1d = act0  + 262144;   // 64*64*16   =  65536
    float* act1  = act1d + 65536;    // 64*64*32   = 131072
    float* act2d = act1  + 131072;   // 32*32*32   =  32768
    float* act2  = act2d + 32768;    // 32*32*64   =  65536
    float* act3d = act2  + 65536;    // 16*16*64   =  16384
    float* act3  = act3d + 16384;    // 16*16*128  =  32768
    float* headb = act3  + 32768;    // 256*16     =   4096

    // Block 0 (fused DW+PW), 256x256x1 -> 128x128x16
    block0_kernel<<<64, 256, 0, stream>>>(
        x, BP(0, 0), BP(0, 1), BP(0, 2), BP(0, 3), BP(0, 4),
        BP(0, 5), BP(0, 6), BP(0, 7), BP(0, 8), BP(0, 9), act0);

    // Block 1: DW 16ch 128->64, PW GEMM M=4096 K=16 N=32 (512 tiles, 4 waves/blk)
    dw_kernel<<<256, 256, 0, stream>>>(act0, BP(1, 0), BP(1, 1), BP(1, 2), BP(1, 3), BP(1, 4),
                                       act1d, 16, 128, 128);
    pw_gemm_wmma<4096, 16, 32><<<128, 128, 0, stream>>>(
        act1d, BP(1, 5), BP(1, 6), BP(1, 7), BP(1, 8), BP(1, 9), act1);

    // Block 2: DW 32ch 64->32, PW GEMM M=1024 K=32 N=64 (256 tiles)
    dw_kernel<<<128, 256, 0, stream>>>(act1, BP(2, 0), BP(2, 1), BP(2, 2), BP(2, 3), BP(2, 4),
                                       act2d, 32, 64, 64);
    pw_gemm_wmma<1024, 32, 64><<<64, 128, 0, stream>>>(
        act2d, BP(2, 5), BP(2, 6), BP(2, 7), BP(2, 8), BP(2, 9), act2);

    // Block 3: DW 64ch 32->16, PW GEMM M=256 K=64 N=128 (128 tiles)
    dw_kernel<<<64, 256, 0, stream>>>(act2, BP(3, 0), BP(3, 1), BP(3, 2), BP(3, 3), BP(3, 4),
                                      act3d, 64, 32, 32);
    pw_gemm_wmma<256, 64, 128><<<32, 128, 0, stream>>>(
        act3d, BP(3, 5), BP(3, 6), BP(3, 7), BP(3, 8), BP(3, 9), act3);

    // Heads (cls+reg) as one GEMM: M=256 K=128 N=16 (16 tiles)
    head_gemm_wmma<<<4, 128, 0, stream>>>(act3, cls_w, cls_b, reg_w, reg_b, headb);

    // Decode + greedy NMS, single workgroup (serial dependency chain)
    decode_nms_kernel<<<1, NANCH, 0, stream>>>(headb, anchors, (float*)d_out);
#undef BP
}